// getDEFeature_45964740002109
// MI455X (gfx1250) — compile-verified
//
#include <hip/hip_runtime.h>
#include <cmath>
#include <complex>

// ---------------- problem constants ----------------
#define ROWS     1024
#define T_LEN    8192
#define PADL     75              // 3 * max(len(a),len(b)) = 3*25
#define EXT_LEN  (T_LEN + 2*PADL)   // 8342
#define NTAP     25
#define NSTATE   24
#define NPERSEG  100
#define SEG_STEP 50
#define NSEG     162             // (8192-100)/50 + 1
#define SEGPAD   176             // 11 * 16
#define NBIN     51              // nperseg/2 + 1
#define MTILES   11
#define NTILES   7               // 112 cols = 2*51 interleaved re/im, padded
#define KSTEPS   25              // 100 / 4
#define ROWPAD   8864            // >= 175*50 + 99 + 1, zero-padded tail

struct Coef { double b[NTAP]; double a[NTAP]; double zi[NSTATE]; };

typedef __attribute__((ext_vector_type(2))) float v2f;
typedef __attribute__((ext_vector_type(8))) float v8f;

// ---------------- kernel 1: 32x32 tiled transpose (row-major -> time-major) ----
__global__ __launch_bounds__(256)
void transpose_kernel(const float* __restrict__ x, float* __restrict__ xT) {
    __shared__ float tile[32][33];
    int bx = blockIdx.x * 32;              // column (time) base
    int by = blockIdx.y * 32;              // row base
    int tx = threadIdx.x & 31;
    int ty = (threadIdx.x >> 5) * 4;       // 8 waves * 4 rows
#pragma unroll
    for (int i = 0; i < 4; ++i)
        tile[ty + i][tx] = x[(size_t)(by + ty + i) * T_LEN + bx + tx];
    __syncthreads();
#pragma unroll
    for (int i = 0; i < 4; ++i)
        xT[(size_t)(bx + ty + i) * ROWS + by + tx] = tile[tx][ty + i];
}

// ---------------- kernel 2: filtfilt (one thread per row, f64 state) ----------
__global__ __launch_bounds__(64)
void filtfilt_kernel(const float* __restrict__ xT, float* __restrict__ yfwd,
                     float* __restrict__ filtT, Coef cf) {
    int row = blockIdx.x * blockDim.x + threadIdx.x;
    if (row >= ROWS) return;

    double rb[NTAP], ra[NTAP];
#pragma unroll
    for (int i = 0; i < NTAP; ++i) { rb[i] = cf.b[i]; ra[i] = cf.a[i]; }

    auto xv = [&](int i) -> double { return (double)xT[(size_t)i * ROWS + row]; };
    auto ext = [&](int i) -> double {           // odd extension, padlen = 75
        if (i < PADL) return 2.0 * xv(0) - xv(PADL - i);
        int j = i - PADL;
        if (j < T_LEN) return xv(j);
        return 2.0 * xv(T_LEN - 1) - xv(T_LEN - 2 - (j - T_LEN));
    };

    double z[NSTATE];
    {   double e0 = ext(0);
#pragma unroll
        for (int i = 0; i < NSTATE; ++i) z[i] = cf.zi[i] * e0;
    }
    // forward pass over extended signal
    for (int t = 0; t < EXT_LEN; ++t) {
        double xt = ext(t);
        double y  = rb[0] * xt + z[0];
#pragma unroll
        for (int i = 0; i < NSTATE - 1; ++i) z[i] = z[i + 1] + xt * rb[i + 1] - y * ra[i + 1];
        z[NSTATE - 1] = xt * rb[NTAP - 1] - y * ra[NTAP - 1];
        yfwd[(size_t)t * ROWS + row] = (float)y;
    }
    // backward pass (filter the reversed sequence, write re-reversed + cropped)
    {   double e0 = (double)yfwd[(size_t)(EXT_LEN - 1) * ROWS + row];
#pragma unroll
        for (int i = 0; i < NSTATE; ++i) z[i] = cf.zi[i] * e0;
    }
    for (int t = EXT_LEN - 1; t >= 0; --t) {
        double xt = (double)yfwd[(size_t)t * ROWS + row];
        double y  = rb[0] * xt + z[0];
#pragma unroll
        for (int i = 0; i < NSTATE - 1; ++i) z[i] = z[i + 1] + xt * rb[i + 1] - y * ra[i + 1];
        z[NSTATE - 1] = xt * rb[NTAP - 1] - y * ra[NTAP - 1];
        int idx = t - PADL;
        if (idx >= 0 && idx < T_LEN) filtT[(size_t)idx * ROWS + row] = (float)y;
    }
}

// ---------------- kernel 3: Welch PSD via WMMA f32 GEMM + entropy -------------
// One block (256 thr = 8 waves) per row.
//   F[seg][col] = sum_k x[seg*50+k] * (win(k)*basis(k,col))  -  mean[seg]*c(col)
// B' = win*basis lives in registers (25 x v2f per lane, built once per job);
// c(col) = sum_k win*basis accumulated during setup. Inner loop: ds_load + wmma.
__global__ __launch_bounds__(256)
void welch_entropy_kernel(const float* __restrict__ filtT, float* __restrict__ out) {
    __shared__ float s_row[ROWPAD];
    __shared__ float s_mean[SEGPAD];
    __shared__ float s_P[64];

    int row = blockIdx.x;
    for (int i = threadIdx.x; i < ROWPAD; i += 256)
        s_row[i] = (i < T_LEN) ? filtT[(size_t)i * ROWS + row] : 0.0f;
    if (threadIdx.x < 64) s_P[threadIdx.x] = 0.0f;
    __syncthreads();

    if (threadIdx.x < SEGPAD) {                    // per-segment mean (detrend)
        float m = 0.0f;
        if (threadIdx.x < NSEG) {
            const float* s = s_row + threadIdx.x * SEG_STEP;
            for (int k = 0; k < NPERSEG; ++k) m += s[k];
            m *= (1.0f / NPERSEG);
        }
        s_mean[threadIdx.x] = m;
    }
    __syncthreads();

    const int wave  = threadIdx.x >> 5;
    const int lane  = threadIdx.x & 31;
    const int mloc  = lane & 15;
    const int kh    = lane >> 4;                   // K half-select per ISA layout
    const int rbase = kh * 8;                      // C/D rows 0-7 or 8-15
    const float W   = 6.2831853071795864769f / (float)NPERSEG;

    // 14 jobs: (n-tile 0..6) x (m-half 0..1), over 8 waves
    for (int job = wave; job < 2 * NTILES; job += 8) {
        int nt   = job >> 1;
        int half = job & 1;
        int mt0  = half ? 6 : 0;
        int mt1  = half ? MTILES : 6;

        int col  = nt * 16 + mloc;                 // B/D column (re/im interleaved)
        int bin  = col >> 1;
        bool isRe = (col & 1) == 0;

        // ---- one-time setup: windowed basis in registers + DC correction ----
        v2f bw[KSTEPS];
        float chalf = 0.0f;
#pragma unroll
        for (int ks = 0; ks < KSTEPS; ++ks) {
            int k0 = ks * 4 + kh * 2;              // even by construction
            float w0 = 0.5f - 0.5f * __cosf(W * (float)k0);
            float w1 = 0.5f - 0.5f * __cosf(W * (float)(k0 + 1));
            float b0 = 0.0f, b1 = 0.0f;
            if (col < 2 * NBIN) {
                float p0 = W * (float)(bin * k0);
                float p1 = W * (float)(bin * (k0 + 1));
                if (isRe) { b0 =  __cosf(p0); b1 =  __cosf(p1); }
                else      { b0 = -__sinf(p0); b1 = -__sinf(p1); }
            }
            v2f t; t.x = b0 * w0; t.y = b1 * w1;
            bw[ks] = t;
            chalf += t.x + t.y;
        }
        float c = chalf + __shfl_xor(chalf, 16, 32);   // combine K halves

        // ---- m-tile loop: pure ds_load_b64 + v_wmma ----
        for (int mt = mt0; mt < mt1; ++mt) {
            int base = (mt * 16 + mloc) * SEG_STEP;    // even -> 8B aligned
            v8f acc = {};
#pragma unroll
            for (int ks = 0; ks < KSTEPS; ++ks) {
                v2f a = *reinterpret_cast<const v2f*>(s_row + base + ks * 4 + kh * 2);
                acc = __builtin_amdgcn_wmma_f32_16x16x4_f32(
                          false, a, false, bw[ks], (short)0, acc, false, false);
            }
            // mean correction + |F|^2, pair re/im via adjacent lanes
            float s = 0.0f;
#pragma unroll
            for (int r = 0; r < 8; ++r) {
                float F = acc[r] - s_mean[mt * 16 + rbase + r] * c;
                s += F * F;
            }
            s += __shfl_xor(s, 1, 32);
            if ((lane & 1) == 0 && bin < NBIN)
                atomicAdd(&s_P[bin], s);               // ds_add_f32
        }
    }
    __syncthreads();

    if (threadIdx.x == 0) {                        // entropy in f64
        const double scale = 1.0 / (1000.0 * 37.5);   // 1/(fs * sum(hann^2))
        double de = 0.0;
        for (int bin = 0; bin < NBIN; ++bin) {
            double P = (double)s_P[bin] * scale / (double)NSEG;
            if (bin > 0 && bin < NBIN - 1) P *= 2.0;
            if (P <= 0.0) P = 2.220446049250313e-16;
            de -= P * log(P);
        }
        out[row] = (float)de;
    }
}

// ---------------- host: Butterworth bandpass design (scipy replica) ----------
static void poly_from_roots(const std::complex<double>* r, int n, std::complex<double>* c) {
    c[0] = std::complex<double>(1.0, 0.0);
    for (int i = 1; i <= n; ++i) c[i] = std::complex<double>(0.0, 0.0);
    for (int i = 0; i < n; ++i)
        for (int j = i + 1; j >= 1; --j) c[j] = c[j] - r[i] * c[j - 1];
}

static Coef make_coef() {
    using C = std::complex<double>;
    const int N = 12;
    C p[12];
    for (int i = 0; i < N; ++i) {
        double m = (double)(-N + 1 + 2 * i);
        p[i] = -std::exp(C(0.0, M_PI * m / (2.0 * N)));
    }
    double wn0 = 80.0 / 500.0, wn1 = 450.0 / 500.0;
    double warped0 = 4.0 * std::tan(M_PI * wn0 / 2.0);
    double warped1 = 4.0 * std::tan(M_PI * wn1 / 2.0);
    double bw = warped1 - warped0;
    double wo = std::sqrt(warped0 * warped1);
    C pbp[24];
    for (int i = 0; i < N; ++i) {
        C plp = p[i] * (bw * 0.5);
        C sq  = std::sqrt(plp * plp - wo * wo);
        pbp[i]     = plp + sq;
        pbp[i + N] = plp - sq;
    }
    double kbp = std::pow(bw, (double)N);
    C pd[24], zd[24], denom(1.0, 0.0);
    for (int i = 0; i < 2 * N; ++i) {
        pd[i] = (4.0 + pbp[i]) / (4.0 - pbp[i]);
        denom *= (4.0 - pbp[i]);
    }
    for (int i = 0; i < N; ++i) { zd[i] = C(1.0, 0.0); zd[i + N] = C(-1.0, 0.0); }
    double kd = kbp * std::real(std::pow(4.0, (double)N) / denom);
    C bc[25], ac[25];
    poly_from_roots(zd, 24, bc);
    poly_from_roots(pd, 24, ac);
    Coef cf;
    for (int i = 0; i < NTAP; ++i) { cf.b[i] = kd * std::real(bc[i]); cf.a[i] = std::real(ac[i]); }
    // lfilter_zi: solve (I - comp^T) zi = b[1:] - a[1:]*b[0]
    double M[NSTATE][NSTATE + 1];
    for (int i = 0; i < NSTATE; ++i) {
        for (int j = 0; j < NSTATE; ++j) {
            double compji = (j == 0) ? -cf.a[i + 1] : ((i == j - 1) ? 1.0 : 0.0);
            M[i][j] = ((i == j) ? 1.0 : 0.0) - compji;
        }
        M[i][NSTATE] = cf.b[i + 1] - cf.a[i + 1] * cf.b[0];
    }
    for (int c0 = 0; c0 < NSTATE; ++c0) {
        int piv = c0;
        for (int r2 = c0 + 1; r2 < NSTATE; ++r2)
            if (std::fabs(M[r2][c0]) > std::fabs(M[piv][c0])) piv = r2;
        if (piv != c0)
            for (int j = c0; j <= NSTATE; ++j) { double t = M[c0][j]; M[c0][j] = M[piv][j]; M[piv][j] = t; }
        double d = M[c0][c0];
        for (int j = c0; j <= NSTATE; ++j) M[c0][j] /= d;
        for (int r2 = 0; r2 < NSTATE; ++r2)
            if (r2 != c0 && M[r2][c0] != 0.0) {
                double f = M[r2][c0];
                for (int j = c0; j <= NSTATE; ++j) M[r2][j] -= f * M[c0][j];
            }
    }
    for (int i = 0; i < NSTATE; ++i) cf.zi[i] = M[i][NSTATE];
    return cf;
}

// ---------------- entry point -----------------------------------------------
extern "C" void kernel_launch(void* const* d_in, const int* in_sizes, int n_in,
                              void* d_out, int out_size, void* d_ws, size_t ws_size,
                              hipStream_t stream) {
    const float* x  = (const float*)d_in[0];
    float* out      = (float*)d_out;

    // workspace layout (time-major):
    //   buf0: [T_LEN][ROWS]  transposed input, later overwritten by filtfilt output
    //   buf1: [EXT_LEN][ROWS] forward-pass intermediate
    float* buf0 = (float*)d_ws;
    float* yfwd = buf0 + (size_t)T_LEN * ROWS;

    Coef cf = make_coef();   // deterministic, recomputed every call

    dim3 tg(T_LEN / 32, ROWS / 32);
    transpose_kernel<<<tg, 256, 0, stream>>>(x, buf0);
    filtfilt_kernel<<<ROWS / 64, 64, 0, stream>>>(buf0, yfwd, buf0, cf);
    welch_entropy_kernel<<<ROWS, 256, 0, stream>>>(buf0, out);
}